// BFA_3015067042007
// MI455X (gfx1250) — compile-verified
//
#include <hip/hip_runtime.h>

#define BATCH 16
#define CH    3
#define H     512
#define Wd    512
#define OC    32
#define HW    (H * Wd)

typedef __attribute__((ext_vector_type(2))) float v2f;
typedef __attribute__((ext_vector_type(8))) float v8f;

// ---------------------------------------------------------------- gray ----
__global__ __launch_bounds__(256) void k_gray(const float* __restrict__ x,
                                              float* __restrict__ gray) {
  int idx = blockIdx.x * 256 + threadIdx.x;
  if (idx >= BATCH * HW) return;
  int b = idx / HW;
  int p = idx - b * HW;
  const float* xb = x + (size_t)b * CH * HW + p;
  float g = 0.2989f * xb[0] + 0.587f * xb[HW] + 0.114f * xb[2 * HW];
  g = fminf(fmaxf(g, 0.0f), 255.0f);
  gray[idx] = floorf(g);
}

// ----------------------------------------------- sobel + NMS + threshold ----
__global__ __launch_bounds__(256) void k_nms(const float* __restrict__ gray,
                                             unsigned char* __restrict__ weak,
                                             unsigned char* __restrict__ cur) {
  __shared__ float gs[36 * 36];   // gray tile, halo 2 (edge-clamped)
  __shared__ float mg[34 * 34];   // |gx|+|gy| tile, halo 1 (zero outside img)
  int blk = blockIdx.x;
  int b = blk >> 8;               // 16x16 tiles of 32x32 per image
  int t = blk & 255;
  int ty0g = (t >> 4) * 32;
  int tx0g = (t & 15) * 32;
  const float* gb = gray + (size_t)b * HW;
  int tid = threadIdx.x;

  for (int i = tid; i < 36 * 36; i += 256) {
    int ly = i / 36, lx = i - ly * 36;
    int y = ty0g + ly - 2, x = tx0g + lx - 2;
    y = min(max(y, 0), H - 1);
    x = min(max(x, 0), Wd - 1);
    gs[i] = gb[y * Wd + x];
  }
  __syncthreads();

  for (int i = tid; i < 34 * 34; i += 256) {
    int ly = i / 34, lx = i - ly * 34;
    int y = ty0g + ly - 1, x = tx0g + lx - 1;
    float m = 0.0f;
    if (y >= 0 && y < H && x >= 0 && x < Wd) {
      const float* c = gs + (ly + 1) * 36 + (lx + 1);
      float gxv = c[-36 + 1] + 2.0f * c[1]  + c[36 + 1]
                - c[-36 - 1] - 2.0f * c[-1] - c[36 - 1];
      float gyv = c[36 - 1] + 2.0f * c[36]  + c[36 + 1]
                - c[-36 - 1] - 2.0f * c[-36] - c[-36 + 1];
      m = fabsf(gxv) + fabsf(gyv);
    }
    mg[i] = m;
  }
  __syncthreads();

  int tx = tid & 31;
  int tyb = tid >> 5;
  size_t base = (size_t)b * HW;
  for (int r = 0; r < 4; ++r) {
    int ty = tyb + r * 8;
    int y = ty0g + ty, x = tx0g + tx;
    const float* c = gs + (ty + 2) * 36 + (tx + 2);
    float gxv = c[-36 + 1] + 2.0f * c[1]  + c[36 + 1]
              - c[-36 - 1] - 2.0f * c[-1] - c[36 - 1];
    float gyv = c[36 - 1] + 2.0f * c[36]  + c[36 + 1]
              - c[-36 - 1] - 2.0f * c[-36] - c[-36 + 1];
    float ax = fabsf(gxv), ay = fabsf(gyv);
    const float* mc = mg + (ty + 1) * 34 + (tx + 1);
    float mag = mc[0];
    float e = mc[1], w = mc[-1], n = mc[-34], s = mc[34];
    float nw = mc[-35], se = mc[35], ne = mc[-33], sw = mc[33];
    bool horiz = ay <= 0.4142135623730951f * ax;
    bool vert  = ay >  2.414213562373095f * ax;
    bool ss    = gxv * gyv >= 0.0f;
    float n1 = horiz ? e : (vert ? n : (ss ? nw : ne));
    float n2 = horiz ? w : (vert ? s : (ss ? se : sw));
    bool keep = (mag > n1) && (mag >= n2);
    float nms = keep ? mag : 0.0f;
    weak[base + y * Wd + x] = nms > 50.0f  ? 1 : 0;
    cur [base + y * Wd + x] = nms > 150.0f ? 1 : 0;
  }
}

// --------------------------------------- hysteresis: tile-local fixpoint ----
__global__ __launch_bounds__(256) void k_hyst(const unsigned char* __restrict__ curin,
                                              const unsigned char* __restrict__ weak,
                                              unsigned char* __restrict__ curout) {
  __shared__ unsigned char cs[34 * 34];
  __shared__ unsigned char wk[34 * 34];
  __shared__ int changed;
  int blk = blockIdx.x;
  int b = blk >> 8, t = blk & 255;
  int ty0g = (t >> 4) * 32, tx0g = (t & 15) * 32;
  size_t base = (size_t)b * HW;
  int tid = threadIdx.x;

  for (int i = tid; i < 34 * 34; i += 256) {
    int ly = i / 34, lx = i - ly * 34;
    int y = ty0g + ly - 1, x = tx0g + lx - 1;
    unsigned char cv = 0, wv = 0;
    if (y >= 0 && y < H && x >= 0 && x < Wd) {
      cv = curin[base + y * Wd + x];
      wv = weak [base + y * Wd + x];
    }
    cs[i] = cv;
    wk[i] = wv;
  }
  __syncthreads();

  int tx = tid & 31, tyb = tid >> 5;
  for (int iter = 0; iter < 1200; ++iter) {   // monotone => terminates <= 1025
    if (tid == 0) changed = 0;
    __syncthreads();
    int ch = 0;
    for (int r = 0; r < 4; ++r) {
      int ty = tyb + r * 8;
      int li = (ty + 1) * 34 + (tx + 1);
      if (!cs[li] && wk[li]) {
        int any = cs[li - 1] | cs[li + 1] | cs[li - 34] | cs[li + 34]
                | cs[li - 35] | cs[li - 33] | cs[li + 33] | cs[li + 35];
        if (any) { cs[li] = 1; ch = 1; }      // benign race, monotone closure
      }
    }
    if (ch) changed = 1;
    __syncthreads();
    int c = changed;
    __syncthreads();                           // all read flag before reset
    if (!c) break;
  }

  for (int r = 0; r < 4; ++r) {
    int ty = tyb + r * 8;
    curout[base + (ty0g + ty) * Wd + (tx0g + tx)] = cs[(ty + 1) * 34 + (tx + 1)];
  }
}

// --------------------------- channel mix: WMMA f32 16x16x4 + bias + ReLU ----
__global__ __launch_bounds__(256) void k_mix(const float* __restrict__ x,
                                             const unsigned char* __restrict__ edge,
                                             const float* __restrict__ Wm,
                                             const float* __restrict__ bias,
                                             float* __restrict__ out) {
  __shared__ float tr[OC * 132];              // 32 channels x 128 px, padded
  int blk = blockIdx.x;
  int b = blk >> 11;                          // 512 rows * 4 segs per image
  int rem = blk & 2047;
  int h = rem >> 2;
  int wbase = (rem & 3) * 128;
  int tid = threadIdx.x;
  int wave = tid >> 5;
  int lane = tid & 31;
  int l16 = lane & 15;
  bool hi = lane >= 16;
  int px = wbase + wave * 16 + l16;           // M = l16 within this wave tile

  // A (16x4): lanes 0-15 -> K={0,1} = {R,G}; lanes 16-31 -> K={2,3} = {B,edge}
  const float* xb = x + (size_t)b * CH * HW + (size_t)h * Wd + px;
  v2f A;
  if (!hi) {
    A.x = xb[0];
    A.y = xb[HW];
  } else {
    A.x = xb[2 * HW];
    A.y = edge[(size_t)b * HW + (size_t)h * Wd + px] ? 255.0f : 0.0f;
  }

  // B (4x16): B[k][n] = W[n][k]; K rows striped like A across half-waves
  int k0 = hi ? 2 : 0;
  v2f B0, B1;
  B0.x = Wm[l16 * 4 + k0];
  B0.y = Wm[l16 * 4 + k0 + 1];
  B1.x = Wm[(l16 + 16) * 4 + k0];
  B1.y = Wm[(l16 + 16) * 4 + k0 + 1];

  v8f acc = {};
  v8f d0 = __builtin_amdgcn_wmma_f32_16x16x4_f32(false, A, false, B0,
                                                 (short)0, acc, false, false);
  v8f d1 = __builtin_amdgcn_wmma_f32_16x16x4_f32(false, A, false, B1,
                                                 (short)0, acc, false, false);

  float bias0 = bias[l16];
  float bias1 = bias[l16 + 16];
  int pixbase = wave * 16 + (hi ? 8 : 0);     // D: lanes16-31 hold M = v+8
#pragma unroll
  for (int v = 0; v < 8; ++v) {
    int pix = pixbase + v;
    tr[l16 * 132 + pix]        = fmaxf(d0[v] + bias0, 0.0f);
    tr[(l16 + 16) * 132 + pix] = fmaxf(d1[v] + bias1, 0.0f);
  }
  __syncthreads();

  // coalesced store: 512B contiguous per channel row segment
  size_t ob = (size_t)b * OC * HW + (size_t)h * Wd + wbase;
  for (int i = tid; i < OC * 128; i += 256) {
    int o = i >> 7, p = i & 127;
    out[ob + (size_t)o * HW + p] = tr[o * 132 + p];
  }
}

// --------------------------------------------------------------- launch ----
extern "C" void kernel_launch(void* const* d_in, const int* in_sizes, int n_in,
                              void* d_out, int out_size, void* d_ws, size_t ws_size,
                              hipStream_t stream) {
  const float* x  = (const float*)d_in[0];
  const float* Wm = (const float*)d_in[1];
  const float* bv = (const float*)d_in[2];
  float* out = (float*)d_out;

  char* ws = (char*)d_ws;
  float*         gray = (float*)ws;                         // 16 MB
  unsigned char* weak = (unsigned char*)(ws + 16777216);    // 4 MB
  unsigned char* cur0 = (unsigned char*)(ws + 20971520);    // 4 MB
  unsigned char* cur1 = (unsigned char*)(ws + 25165824);    // 4 MB

  k_gray<<<(BATCH * HW + 255) / 256, 256, 0, stream>>>(x, gray);
  k_nms<<<BATCH * 256, 256, 0, stream>>>(gray, weak, cur0);

  // 20 ping-pong passes; each pass closes to the local tile fixpoint, so
  // cross-tile propagation is ~a full tile per pass (vs 1 px in the reference)
  for (int pass = 0; pass < 20; ++pass) {
    const unsigned char* ci = (pass & 1) ? cur1 : cur0;
    unsigned char*       co = (pass & 1) ? cur0 : cur1;
    k_hyst<<<BATCH * 256, 256, 0, stream>>>(ci, weak, co);
  }
  // after 20 passes the result is back in cur0

  k_mix<<<BATCH * 2048, 256, 0, stream>>>(x, cur0, Wm, bv, out);
}